// VoxelQueryAndGroup_54898271977787
// MI455X (gfx1250) — compile-verified
//
#include <hip/hip_runtime.h>

typedef __attribute__((ext_vector_type(2))) float v2f;
typedef __attribute__((ext_vector_type(8))) float v8f;

#define BB 4
#define NN 16384
#define MM 2048
#define CC 64
#define SS 32
// MAX_RADIUS^2 = 0.04f = 0x3D23D70A. MIN_RADIUS == 0.
// (d2 >= 0 && d2 < 0.04f)  <=>  bit_cast<uint>(d2) < 0x3D23D70A  (d2 never NaN):
// non-negative floats order like their bit patterns; negatives compare huge.
#define R2MAX_BITS 0x3D23D70Au

#if __has_builtin(__builtin_amdgcn_ballot_w32)
#define BALLOT32(p) __builtin_amdgcn_ballot_w32(p)
#else
#define BALLOT32(p) ((unsigned)__ballot(p))
#endif

// ---------------------------------------------------------------------------
// Kernel 1: ball query. One wave (wave32) owns 16 centers. Per 16-point tile:
//   1 x V_WMMA_F32_16X16X4_F32 computes the 16x16 d^2 tile via the
//   homogeneous trick: A row m = [-2cx,-2cy,-2cz,1], B col n = [px,py,pz,|p|^2],
//   C init = |c|^2  =>  D = |c|^2 + |p|^2 - 2 c.p.
// Then 8 wave-wide ballots (each a single v_cmp_lt_u32, folded straight into
// the ballot mask) turn the tile into 16 per-center hit masks
// (D layout: lane l = col n=l&15, rows m=v+8*(l>=16), so ballot(v) bits[15:0]
// = center v, bits[31:16] = center v+8). Lanes 0..15 pop bits in ascending n
// (matching the reference's "32 smallest candidate indices" semantics), with
// wave-uniform early exit once every center has 32 hits (~6% of N scanned on
// average for uniform data).
// ---------------------------------------------------------------------------
__global__ __launch_bounds__(256) void ball_query_wmma_kernel(
    const float* __restrict__ points,    // [B,N,3]
    const float* __restrict__ centers,   // [B,M,3]
    int* __restrict__ idx_out)           // [B,M,SS]
{
  const int lane = threadIdx.x & 31;
  const int half = lane >> 4;            // 0: lanes 0-15 (K=0,1), 1: lanes 16-31 (K=2,3)
  const int l16  = lane & 15;
  const int moff = half * 8;

  const int gwave  = blockIdx.x * (blockDim.x >> 5) + (threadIdx.x >> 5);
  const int b      = gwave / (MM / 16);
  const int m_base = (gwave % (MM / 16)) * 16;

  // A operand (16x4 MxK, f32): lane<16 holds K=0,1 of row M=lane,
  // lane>=16 holds K=2,3 of row M=lane-16.
  const float* cp = centers + ((size_t)b * MM + m_base + l16) * 3;
  const float cx = cp[0], cy = cp[1], cz = cp[2];
  v2f A;
  A.x = half ? (-2.0f * cz) : (-2.0f * cx);
  A.y = half ? 1.0f         : (-2.0f * cy);

  // C init: C[m][n] = |c_m|^2. Every lane knows |c|^2 of center l16; shuffle
  // the rows it needs (m = moff + v). LDS-free (8 one-time ds_bpermute).
  const float c2 = cx * cx + cy * cy + cz * cz;
  v8f Cin;
  #pragma unroll
  for (int v = 0; v < 8; ++v) Cin[v] = __shfl(c2, moff + v, 32);

  int cnt   = (lane < 16) ? 0 : SS;      // upper lanes vacuously "full" for __all
  int first = -1;
  int* orow = idx_out + ((size_t)b * MM + m_base + l16) * SS;
  const int vsel = lane & 7;
  const bool hi_half = (lane & 8) != 0;

  // Strength-reduced point pointer: advance 16 points (192 B) per tile.
  const float* pp = points + ((size_t)b * NN + l16) * 3;

  for (int nt = 0; nt < NN / 16; ++nt, pp += 16 * 3) {
    if (__all(cnt >= SS)) break;         // uniform early exit: all centers full

    const int n0 = nt * 16;
    // B operand (4x16 KxN, f32): lane<16 holds K=0,1 of col N=lane,
    // lane>=16 holds K=2,3 of col N=lane-16.
    const float px = pp[0], py = pp[1], pz = pp[2];
    v2f Bv;
    Bv.x = half ? pz : px;
    Bv.y = half ? (px * px + py * py + pz * pz) : py;

    v8f D = __builtin_amdgcn_wmma_f32_16x16x4_f32(
        false, A, false, Bv, (short)0, Cin, false, false);

    // 8 single-compare wave32 ballots -> 16 per-center hit masks.
    unsigned bal[8];
    #pragma unroll
    for (int v = 0; v < 8; ++v)
      bal[v] = BALLOT32(__float_as_uint(D[v]) < R2MAX_BITS);

    // Scan lane m (=lane, 0..15) selects ballot v = m&7, half by m>=8.
    unsigned mb = bal[0];
    #pragma unroll
    for (int v = 1; v < 8; ++v) mb = (vsel == v) ? bal[v] : mb;
    unsigned bits = hi_half ? (mb >> 16) : (mb & 0xffffu);

    // Pop hits in ascending point order; expected ~0.5 iterations per tile.
    while (bits != 0u && cnt < SS) {
      const int n = n0 + __builtin_ctz(bits);
      bits &= bits - 1u;
      if (cnt == 0) first = n;
      orow[cnt] = n;
      ++cnt;
    }
  }

  // Backfill: empty slots get first hit; no hits at all -> 0 (matches reference)
  if (lane < 16) {
    const int fill = (first < 0) ? 0 : first;
    for (int s = cnt; s < SS; ++s) orow[s] = fill;
  }
}

// ---------------------------------------------------------------------------
// Kernel 2: grouping gather. Thread per (b, m, s); writes along s are
// contiguous per wave (coalesced). Features (16.8 MB) live in the 192 MB L2,
// so the random gather is L2-bandwidth bound; the 70 MB streamed output uses
// non-temporal stores so it does not evict the gather working set.
// out: [B, 3+C, M, SS]
// ---------------------------------------------------------------------------
__global__ __launch_bounds__(256) void group_gather_kernel(
    const float* __restrict__ points,    // [B,N,3]
    const float* __restrict__ centers,   // [B,M,3]
    const float* __restrict__ features,  // [B,C,N]
    const int* __restrict__ idx,         // [B,M,SS]
    float* __restrict__ out)             // [B,3+C,M,SS]
{
  const int t = blockIdx.x * blockDim.x + threadIdx.x;  // B*M*SS threads
  const int s = t & (SS - 1);
  const int m = (t >> 5) & (MM - 1);
  const int b = t >> 16;                                // 5 + log2(MM=2048)

  const int id = idx[t];
  const float* p = points  + ((size_t)b * NN + id) * 3;
  const float* c = centers + ((size_t)b * MM + m) * 3;

  const size_t chstride = (size_t)MM * SS;
  size_t o = (size_t)b * (3 + CC) * chstride + (size_t)m * SS + s;

  __builtin_nontemporal_store(p[0] - c[0], &out[o]);
  __builtin_nontemporal_store(p[1] - c[1], &out[o + chstride]);
  __builtin_nontemporal_store(p[2] - c[2], &out[o + 2 * chstride]);
  o += 3 * chstride;

  const float* f = features + (size_t)b * CC * NN + id;
  #pragma unroll 4
  for (int ch = 0; ch < CC; ++ch)
    __builtin_nontemporal_store(f[(size_t)ch * NN], &out[o + (size_t)ch * chstride]);
}

// ---------------------------------------------------------------------------
extern "C" void kernel_launch(void* const* d_in, const int* in_sizes, int n_in,
                              void* d_out, int out_size, void* d_ws, size_t ws_size,
                              hipStream_t stream) {
  const float* points   = (const float*)d_in[0];  // (B,N,3)
  const float* centers  = (const float*)d_in[1];  // (B,M,3)
  const float* features = (const float*)d_in[2];  // (B,C,N)
  float* out = (float*)d_out;                     // (B,3+C,M,SS)
  int* idx = (int*)d_ws;                          // B*M*SS ints = 1 MiB scratch

  // 512 waves total (one per 16 centers), 8 waves / 256-thread block
  ball_query_wmma_kernel<<<(BB * (MM / 16)) / 8, 256, 0, stream>>>(points, centers, idx);
  // one thread per (b, m, s)
  group_gather_kernel<<<(BB * MM * SS) / 256, 256, 0, stream>>>(points, centers, features, idx, out);
}